// SAGEPL_40003325394971
// MI455X (gfx1250) — compile-verified
//
#include <hip/hip_runtime.h>
#include <hip/hip_bf16.h>

// ---------------------------------------------------------------------------
// GraphSAGE 3-layer stack, run twice (clean + noisy), fp32 end-to-end.
//  - GEMMs: V_WMMA_F32_16X16X4_F32 (CDNA5 fp32 matrix pipe, matches fp32 ref)
//  - Operand staging: Tensor Data Mover (tensor_load_to_lds, 6-arg clang-23
//    form) global->LDS with D#-descriptor padding for bank spread and OOB
//    zero-fill; ds_load in the inner loop. Cooperative-load fallback if the
//    builtin is absent.
//  - Aggregation: edge-parallel f32 atomics (activations ~51MB fit in 192MB L2)
// ---------------------------------------------------------------------------

typedef __attribute__((ext_vector_type(2))) float v2f;
typedef __attribute__((ext_vector_type(8))) float v8f;
typedef __attribute__((ext_vector_type(4))) unsigned int u32x4;
typedef __attribute__((ext_vector_type(8))) int          i32x8;
typedef __attribute__((ext_vector_type(4))) int          i32x4;

#define SAGE_IN   128
#define SAGE_H    256
#define SAGE_OUT  47

#define SA_STRIDE 34   // 32 K floats + 2 pad  (TDM: pad_interval=4, pad_amount=1)
#define SB_STRIDE 80   // 64 N floats + 16 pad (TDM: pad_interval=5, pad_amount=15)

#if defined(__HIP_DEVICE_COMPILE__) && __has_builtin(__builtin_amdgcn_tensor_load_to_lds) && \
    __has_builtin(__builtin_amdgcn_s_wait_tensorcnt)
#define SAGE_TDM 1
#else
#define SAGE_TDM 0
#endif

static __device__ __forceinline__ v8f wmma_f32(v2f a, v2f b, v8f c) {
  // (neg_a, A, neg_b, B, c_mod, C, reuse_a, reuse_b)
  return __builtin_amdgcn_wmma_f32_16x16x4_f32(false, a, false, b, (short)0, c,
                                               false, false);
}

#if SAGE_TDM
// Issue one TDM 2D tile load: global (row-major, stride0 elems of 4B) -> LDS.
// tile_d0 = contiguous dim length, tile_d1 = #rows. OOB rows/cols zero-fill.
// pad_interval/pad_amount (ISA D# encoding) insert LDS padding after each row.
static __device__ __forceinline__ void tdm_load_2d(const float* gptr,
                                                   unsigned lds_byte_addr,
                                                   unsigned tensor_d0,
                                                   unsigned tensor_d1,
                                                   unsigned stride0_elems,
                                                   unsigned tile_d0,
                                                   unsigned tile_d1,
                                                   unsigned pad_interval,
                                                   unsigned pad_amount) {
  unsigned long long ga = (unsigned long long)(__UINTPTR_TYPE__)gptr;
  u32x4 g0;
  g0.x = 1u;                                   // count=1, user descriptor
  g0.y = lds_byte_addr;                        // LDS byte address of tile
  g0.z = (unsigned)(ga & 0xFFFFFFFFu);         // global_addr[31:0]
  g0.w = (unsigned)(ga >> 32) | (2u << 30);    // global_addr[56:32] | type=2

  i32x8 g1;
  unsigned w0 = (2u << 16)                     // data_size = 2 -> 4 bytes
              | (1u << 20)                     // pad_enable
              | (pad_interval << 22)
              | (pad_amount << 25);
  g1[0] = (int)w0;
  g1[1] = (int)((tensor_d0 & 0xFFFFu) << 16);                       // dim0 lo
  g1[2] = (int)((tensor_d0 >> 16) | ((tensor_d1 & 0xFFFFu) << 16)); // dim0 hi | dim1 lo
  g1[3] = (int)((tensor_d1 >> 16) | (tile_d0 << 16));               // dim1 hi | tile0
  g1[4] = (int)(tile_d1 & 0xFFFFu);                                 // tile1 | tile2=0
  g1[5] = (int)stride0_elems;                                       // dim0_stride lo32
  g1[6] = 0;                                                        // stride hi | d1stride lo
  g1[7] = 0;
  i32x4 z4 = {0, 0, 0, 0};
  i32x8 z8 = {0, 0, 0, 0, 0, 0, 0, 0};
  // clang-23 / therock-10.0 6-arg form: (g0, g1, g2, g3, g4, cpol)
  __builtin_amdgcn_tensor_load_to_lds(g0, g1, z4, z4, z8, 0);
}
#endif

// --------------------------- utility kernels -------------------------------

__global__ void k_zero_f32(float* __restrict__ p, long long n) {
  long long i = (long long)blockIdx.x * blockDim.x + threadIdx.x;
  if (i < n) p[i] = 0.0f;
}

__global__ void k_count_deg(float* __restrict__ deg, const int* __restrict__ dst,
                            int E) {
  int e = blockIdx.x * blockDim.x + threadIdx.x;
  if (e < E) atomicAdd(&deg[dst[e]], 1.0f);
}

__global__ void k_finalize_deg(float* __restrict__ deg, int N) {
  int i = blockIdx.x * blockDim.x + threadIdx.x;
  if (i < N) {
    float d = deg[i];
    deg[i] = 1.0f / (d < 1.0f ? 1.0f : d);
  }
}

// noisy_x = x + sign(x) * (noise / max(||noise||_2, eps)) * 0.3 ; wave per row
__global__ void k_noisy(const float* __restrict__ x,
                        const float* __restrict__ noise,
                        float* __restrict__ out, int N) {
  const int lane = threadIdx.x & 31;
  const int wave = threadIdx.x >> 5;
  const int row  = blockIdx.x * 8 + wave;
  if (row >= N) return;
  const float* nr = noise + (long long)row * SAGE_IN;
  float s = 0.0f;
#pragma unroll
  for (int j = lane; j < SAGE_IN; j += 32) { float v = nr[j]; s += v * v; }
#pragma unroll
  for (int m = 16; m; m >>= 1) s += __shfl_xor(s, m, 32);
  float nrm = sqrtf(s);
  nrm = nrm > 1e-12f ? nrm : 1e-12f;
  const float scl = 0.3f / nrm;
  const float* xr = x + (long long)row * SAGE_IN;
  float* o = out + (long long)row * SAGE_IN;
#pragma unroll
  for (int j = lane; j < SAGE_IN; j += 32) {
    float xv = xr[j];
    float sg = (xv > 0.0f) ? 1.0f : ((xv < 0.0f) ? -1.0f : 0.0f);
    o[j] = xv + sg * nr[j] * scl;
  }
}

// one block per edge; feature dim coalesced; f32 global atomics (L2-resident)
__global__ void k_scatter_add(float* __restrict__ agg,
                              const float* __restrict__ X,
                              const int* __restrict__ src,
                              const int* __restrict__ dst,
                              int din) {
  const int e = blockIdx.x;
  const int s = src[e];
  const int d = dst[e];
  const float* xs = X + (long long)s * din;
  float* ad = agg + (long long)d * din;
  for (int f = threadIdx.x; f < din; f += blockDim.x)
    atomicAdd(&ad[f], xs[f]);
}

// --------------------------- WMMA SAGE GEMM --------------------------------
// Out[rows,dout] = (Xa * inv_deg[row]) @ Wl + bl + Xr @ Wr   (+ optional relu)
// Block = 256 threads = 8 waves -> 128 rows x 64 cols. Per 32-wide K chunk,
// wave 0 DMAs A (128x32) and B (32x64) tiles into padded LDS via the TDM
// (OOB zero-fill handles ragged dout / last row block), then every wave runs
// 8 fully-unrolled f32 WMMA steps off ds_load. EXEC is all-1s at every WMMA.
__global__ void __launch_bounds__(256)
k_sage_gemm(const float* __restrict__ Xa, const float* __restrict__ inv_deg,
            const float* __restrict__ Wl, const float* __restrict__ bl,
            const float* __restrict__ Xr, const float* __restrict__ Wr,
            float* __restrict__ Out, int rows, int din, int dout, int do_relu) {
  __shared__ float sA[128 * SA_STRIDE];
  __shared__ float sB[32 * SB_STRIDE];

  const int lane = threadIdx.x & 31;
  const int wave = threadIdx.x >> 5;
  const int half = lane >> 4;          // 0: K {0,1}, rows 0-7 ; 1: K {2,3}, rows 8-15
  const int l16  = lane & 15;

  const int row0b = blockIdx.x * 128;
  const int row0  = row0b + wave * 16;
  const int col0  = blockIdx.y * 64;
  if (row0 >= rows) return;            // uniform per-wave exit; terminated waves
                                       // auto-satisfy CDNA5 barrier membership

  v8f acc[4];
#pragma unroll
  for (int t = 0; t < 4; ++t) acc[t] = (v8f){0, 0, 0, 0, 0, 0, 0, 0};

  const int   arow   = row0 + l16;     // A row consumed by this lane
  const float dscale = inv_deg[arow];

  for (int pass = 0; pass < 2; ++pass) {
    const float* __restrict__ X = pass ? Xr : Xa;
    const float* __restrict__ W = pass ? Wr : Wl;
    const float s = pass ? 1.0f : dscale;

    for (int kc = 0; kc < din; kc += 32) {
      // ---- stage tiles into LDS ----
#if SAGE_TDM
      if (wave == 0) {
        // B: rows = K (32), contiguous dim = columns (64-wide tile, stride 80)
        tdm_load_2d(W + (long long)kc * dout + col0,
                    (unsigned)(__UINTPTR_TYPE__)&sB[0],
                    (unsigned)(dout - col0), (unsigned)(din - kc),
                    (unsigned)dout, 64u, 32u, /*pad 64DW->+16DW*/ 5u, 15u);
        // A: rows = node rows (128), contiguous dim = K (32-wide, stride 34)
        tdm_load_2d(X + (long long)row0b * din + kc,
                    (unsigned)(__UINTPTR_TYPE__)&sA[0],
                    (unsigned)(din - kc), (unsigned)(rows - row0b),
                    (unsigned)din, 32u, 128u, /*pad 32DW->+2DW*/ 4u, 1u);
        __builtin_amdgcn_s_wait_tensorcnt(0);
      }
#else
      for (int idx = threadIdx.x; idx < 32 * 64; idx += 256) {
        int r = idx >> 6, c = idx & 63;
        int col = col0 + c;
        sB[r * SB_STRIDE + c] =
            (col < dout) ? W[(long long)(kc + r) * dout + col] : 0.0f;
      }
      for (int idx = threadIdx.x; idx < 128 * 32; idx += 256) {
        int r = idx >> 5, c = idx & 31;
        int row = row0b + r;
        sA[r * SA_STRIDE + c] =
            (row < rows) ? X[(long long)row * din + kc + c] : 0.0f;
      }
#endif
      __syncthreads();

      // ---- 8 x WMMA K-steps off LDS ----
      const float* aBase = &sA[(wave * 16 + l16) * SA_STRIDE + 2 * half];
#pragma unroll
      for (int kk = 0; kk < 32; kk += 4) {
        v2f a = *(const v2f*)(aBase + kk);     // ds_load_b64 (8B aligned)
        a.x *= s;
        a.y *= s;
        const float* b0 = &sB[(kk + 2 * half + 0) * SB_STRIDE + l16];
        const float* b1 = &sB[(kk + 2 * half + 1) * SB_STRIDE + l16];
#pragma unroll
        for (int t = 0; t < 4; ++t) {
          v2f b;
          b.x = b0[t * 16];
          b.y = b1[t * 16];
          acc[t] = wmma_f32(a, b, acc[t]);
        }
      }
      __syncthreads();                         // LDS reuse next chunk
    }
  }

  // ---- epilogue: bias + relu + store (divergence after all WMMAs) ----
#pragma unroll
  for (int t = 0; t < 4; ++t) {
    int col = col0 + t * 16 + l16;
    if (col >= dout) continue;
    float bv = bl[col];
#pragma unroll
    for (int r = 0; r < 8; ++r) {
      int row = row0 + r + 8 * half;           // D: VGPR r -> M = r + 8*half
      float v = acc[t][r] + bv;
      if (do_relu) v = v > 0.0f ? v : 0.0f;
      Out[(long long)row * dout + col] = v;
    }
  }
}

// log_softmax over 47 classes: one wave32 per row
__global__ void k_logsoftmax(const float* __restrict__ X, float* __restrict__ out,
                             int N, int C) {
  const int lane = threadIdx.x & 31;
  const int wave = threadIdx.x >> 5;
  const int row  = blockIdx.x * 8 + wave;
  if (row >= N) return;
  const float* xr = X + (long long)row * C;
  float mx = -3.402823466e38f;
  for (int j = lane; j < C; j += 32) mx = fmaxf(mx, xr[j]);
#pragma unroll
  for (int m = 16; m; m >>= 1) mx = fmaxf(mx, __shfl_xor(mx, m, 32));
  float s = 0.0f;
  for (int j = lane; j < C; j += 32) s += expf(xr[j] - mx);
#pragma unroll
  for (int m = 16; m; m >>= 1) s += __shfl_xor(s, m, 32);
  const float lse = mx + logf(s);
  float* o = out + (long long)row * C;
  for (int j = lane; j < C; j += 32) o[j] = xr[j] - lse;
}

// ------------------------------ launcher -----------------------------------

extern "C" void kernel_launch(void* const* d_in, const int* in_sizes, int n_in,
                              void* d_out, int out_size, void* d_ws, size_t ws_size,
                              hipStream_t stream) {
  const float* x     = (const float*)d_in[0];
  const int*   ei    = (const int*)d_in[1];
  const float* noise = (const float*)d_in[2];
  const float* Wl[3] = {(const float*)d_in[3], (const float*)d_in[6], (const float*)d_in[9]};
  const float* bl[3] = {(const float*)d_in[4], (const float*)d_in[7], (const float*)d_in[10]};
  const float* Wr[3] = {(const float*)d_in[5], (const float*)d_in[8], (const float*)d_in[11]};

  const int N = in_sizes[0] / SAGE_IN;   // 50000
  const int E = in_sizes[1] / 2;         // 800000
  const int* src = ei;
  const int* dst = ei + E;

  // outputs, concatenated flat in return order
  float* out  = (float*)d_out;
  float* h_p  = out;
  float* ls_p = h_p  + (size_t)N * SAGE_H;
  float* x_p  = ls_p + (size_t)N * SAGE_OUT;
  float* h_n  = x_p  + (size_t)N * SAGE_OUT;
  float* ls_n = h_n  + (size_t)N * SAGE_H;
  float* x_n  = ls_n + (size_t)N * SAGE_OUT;

  // workspace layout (floats)
  float* ws      = (float*)d_ws;
  float* inv_deg = ws;                              // N
  float* agg     = inv_deg + N;                     // N*256
  float* buf1    = agg + (size_t)N * SAGE_H;        // N*256 (layer-0 output)
  float* noisy   = buf1 + (size_t)N * SAGE_H;       // N*128

  const int B = 256;
  // ---- degree (shared by all 6 aggregations) ----
  k_zero_f32<<<(N + B - 1) / B, B, 0, stream>>>(inv_deg, (long long)N);
  k_count_deg<<<(E + B - 1) / B, B, 0, stream>>>(inv_deg, dst, E);
  k_finalize_deg<<<(N + B - 1) / B, B, 0, stream>>>(inv_deg, N);
  // ---- noisy input ----
  k_noisy<<<(N + 7) / 8, B, 0, stream>>>(x, noise, noisy, N);

  const float* Xin[2]  = {x, noisy};
  float* Hout[2]  = {h_p, h_n};
  float* LSout[2] = {ls_p, ls_n};
  float* Xout[2]  = {x_p, x_n};

  const dim3 gemm_grid((N + 127) / 128, (SAGE_H + 63) / 64);     // 391 x 4
  const dim3 gemm_grid_o((N + 127) / 128, (SAGE_OUT + 63) / 64); // 391 x 1

  for (int s2 = 0; s2 < 2; ++s2) {
    const float* X0 = Xin[s2];

    // layer 0: 128 -> 256, relu, -> buf1
    k_zero_f32<<<((long long)N * SAGE_IN + B - 1) / B, B, 0, stream>>>(
        agg, (long long)N * SAGE_IN);
    k_scatter_add<<<E, 128, 0, stream>>>(agg, X0, src, dst, SAGE_IN);
    k_sage_gemm<<<gemm_grid, B, 0, stream>>>(agg, inv_deg, Wl[0], bl[0], X0,
                                             Wr[0], buf1, N, SAGE_IN, SAGE_H, 1);

    // layer 1: 256 -> 256, relu, -> h output region (also layer-2 input)
    k_zero_f32<<<((long long)N * SAGE_H + B - 1) / B, B, 0, stream>>>(
        agg, (long long)N * SAGE_H);
    k_scatter_add<<<E, 256, 0, stream>>>(agg, buf1, src, dst, SAGE_H);
    k_sage_gemm<<<gemm_grid, B, 0, stream>>>(agg, inv_deg, Wl[1], bl[1], buf1,
                                             Wr[1], Hout[s2], N, SAGE_H, SAGE_H, 1);

    // layer 2: 256 -> 47, no relu, -> x output region
    k_zero_f32<<<((long long)N * SAGE_H + B - 1) / B, B, 0, stream>>>(
        agg, (long long)N * SAGE_H);
    k_scatter_add<<<E, 256, 0, stream>>>(agg, Hout[s2], src, dst, SAGE_H);
    k_sage_gemm<<<gemm_grid_o, B, 0, stream>>>(agg, inv_deg, Wl[2], bl[2],
                                               Hout[s2], Wr[2], Xout[s2], N,
                                               SAGE_H, SAGE_OUT, 0);

    // log_softmax
    k_logsoftmax<<<(N + 7) / 8, B, 0, stream>>>(Xout[s2], LSout[s2], N, SAGE_OUT);
  }
}